// MultiBandSparseModel_81492709474776
// MI455X (gfx1250) — compile-verified
//
#include <hip/hip_runtime.h>
#include <float.h>
#include <math.h>

// ---------------------------------------------------------------------------
// MultiBandSparseModel for MI455X (gfx1250, wave32).
// fp32 end-to-end; all GEMM-like convs use V_WMMA_F32_16X16X4_F32 with
//  - zero-padded operand buffers (no guards -> EXEC stays all-ones, no
//    exec-mask save/restore around WMMA),
//  - 4 independent accumulators per wave (4 N-tiles, shared A fragment) to
//    hide the WMMA->WMMA RAW hazard latency,
//  - global_prefetch_b8 of upcoming K-rows.
// DFT-based band split/merge (exact phase via power-of-two modular index).
// Deterministic: no float atomics, fixed reduction orders.
// Workspace use: ~65 MB.
// ---------------------------------------------------------------------------

typedef float v2f __attribute__((ext_vector_type(2)));
typedef float v8f __attribute__((ext_vector_type(8)));

#define BATCH   4
#define NSAMP   32768
#define NFREQ   16384          // full-spectrum indices used: [0, 16384)
#define TWO_PI  6.283185307179586f

// ------------------------------ DFT kernels --------------------------------

// C[b,k] = (1/sqrt(N)) * sum_t x[b,t] * e^{-2pi i k t / N},  k in [0,16384)
__global__ void dft_forward_kernel(const float* __restrict__ x,
                                   float* __restrict__ Cre, float* __restrict__ Cim) {
    int k = blockIdx.x;
    int b = blockIdx.y;
    const float* xb = x + (size_t)b * NSAMP;
    float sre = 0.f, sim = 0.f;
    for (int t = threadIdx.x; t < NSAMP; t += blockDim.x) {
        unsigned m = ((unsigned)k * (unsigned)t) & (NSAMP - 1);
        float ang = -TWO_PI * (float)m / (float)NSAMP;
        float s, c; __sincosf(ang, &s, &c);
        float xv = xb[t];
        sre += xv * c;
        sim += xv * s;
    }
    __shared__ float rs[256], is_[256];
    rs[threadIdx.x] = sre; is_[threadIdx.x] = sim;
    __syncthreads();
    for (int off = 128; off > 0; off >>= 1) {
        if (threadIdx.x < off) {
            rs[threadIdx.x] += rs[threadIdx.x + off];
            is_[threadIdx.x] += is_[threadIdx.x + off];
        }
        __syncthreads();
    }
    if (threadIdx.x == 0) {
        float sc = rsqrtf((float)NSAMP);
        Cre[b * NFREQ + k] = rs[0] * sc;
        Cim[b * NFREQ + k] = is_[0] * sc;
    }
}

// bandPad[b][64 + t] = (1/sqrt(bs)) * sum_{k in [kstart,kstop)} w_k*(Cre cos - Cim sin)
__global__ void band_synth_kernel(const float* __restrict__ Cre, const float* __restrict__ Cim,
                                  float* __restrict__ bandPad, int bs, int kstart, int kstop) {
    int t = blockIdx.x;
    int b = blockIdx.y;
    const float* cr = Cre + (size_t)b * NFREQ;
    const float* ci = Cim + (size_t)b * NFREQ;
    int nk = kstop - kstart;
    float acc = 0.f;
    for (int kk = threadIdx.x; kk < nk; kk += blockDim.x) {
        int k = kstart + kk;
        unsigned m = ((unsigned)k * (unsigned)t) & (unsigned)(bs - 1);
        float ang = TWO_PI * (float)m / (float)bs;
        float s, c; __sincosf(ang, &s, &c);
        float w = (k == 0) ? 1.f : 2.f;
        acc += w * (cr[k] * c - ci[k] * s);
    }
    __shared__ float sd[256];
    sd[threadIdx.x] = acc; __syncthreads();
    for (int off = 128; off > 0; off >>= 1) {
        if (threadIdx.x < off) sd[threadIdx.x] += sd[threadIdx.x + off];
        __syncthreads();
    }
    if (threadIdx.x == 0) bandPad[(size_t)b * (bs + 128) + 64 + t] = sd[0] * rsqrtf((float)bs);
}

// T[b,k] = (1/sqrt(bs)) * sum_t recon[b,t] e^{-2pi i k t / bs}, k in [kstart,kstop)
__global__ void band_dft_kernel(const float* __restrict__ recon,
                                float* __restrict__ Tre, float* __restrict__ Tim,
                                int bs, int kstart) {
    int k = kstart + blockIdx.x;
    int b = blockIdx.y;
    const float* rb = recon + (size_t)b * bs;
    float sre = 0.f, sim = 0.f;
    for (int t = threadIdx.x; t < bs; t += blockDim.x) {
        unsigned m = ((unsigned)k * (unsigned)t) & (unsigned)(bs - 1);
        float ang = -TWO_PI * (float)m / (float)bs;
        float s, c; __sincosf(ang, &s, &c);
        float rv = rb[t];
        sre += rv * c;
        sim += rv * s;
    }
    __shared__ float rs[256], is_[256];
    rs[threadIdx.x] = sre; is_[threadIdx.x] = sim;
    __syncthreads();
    for (int off = 128; off > 0; off >>= 1) {
        if (threadIdx.x < off) {
            rs[threadIdx.x] += rs[threadIdx.x + off];
            is_[threadIdx.x] += is_[threadIdx.x + off];
        }
        __syncthreads();
    }
    if (threadIdx.x == 0) {
        float sc = rsqrtf((float)bs);
        Tre[b * NFREQ + k] = rs[0] * sc;
        Tim[b * NFREQ + k] = is_[0] * sc;
    }
}

// out[b,t] = (1/sqrt(N)) * sum_k w_k*(Tre cos - Tim sin)
__global__ void final_idft_kernel(const float* __restrict__ Tre, const float* __restrict__ Tim,
                                  float* __restrict__ out) {
    int t = blockIdx.x;
    int b = blockIdx.y;
    const float* tr = Tre + (size_t)b * NFREQ;
    const float* ti = Tim + (size_t)b * NFREQ;
    float acc = 0.f;
    for (int k = threadIdx.x; k < NFREQ; k += blockDim.x) {
        unsigned m = ((unsigned)k * (unsigned)t) & (NSAMP - 1);
        float ang = TWO_PI * (float)m / (float)NSAMP;
        float s, c; __sincosf(ang, &s, &c);
        float w = (k == 0) ? 1.f : 2.f;
        acc += w * (tr[k] * c - ti[k] * s);
    }
    __shared__ float sd[256];
    sd[threadIdx.x] = acc; __syncthreads();
    for (int off = 128; off > 0; off >>= 1) {
        if (threadIdx.x < off) sd[threadIdx.x] += sd[threadIdx.x + off];
        __syncthreads();
    }
    if (threadIdx.x == 0) out[(size_t)b * NSAMP + t] = sd[0] * rsqrtf((float)NSAMP);
}

// ---------------------------- pad-zero kernels -----------------------------

__global__ void zero_band_pad_kernel(float* __restrict__ bandPad, int bs) {
    int b = blockIdx.x;                 // BATCH
    int j = threadIdx.x;                // 128
    float* p = bandPad + (size_t)b * (bs + 128);
    int idx = (j < 64) ? j : (bs + 64 + (j - 64));
    p[idx] = 0.f;
}

// 32 channels x 8 pad elements, stride bs+8, halo of 4 each side
__global__ void zero_ch_pad_kernel(float* __restrict__ buf, int bs) {
    int tid = threadIdx.x;              // 256
    int o = tid >> 3, j = tid & 7;
    int idx = o * (bs + 8) + ((j < 4) ? j : (bs + 4 + (j - 4)));
    buf[idx] = 0.f;
}

// ------------------------------ WMMA kernels -------------------------------
// V_WMMA_F32_16X16X4_F32, one wave per block (blockDim=32), 4 N-tiles/wave.
// A layout: lane = (half<<4)|mr, a[v] = A[M=mr][K = k0 + 2*half + v]
// B layout (mirror): b[v] = B[K = k0 + 2*half + v][N = mr]
// D layout: acc[r] -> D[M = r + 8*half][N = mr]

// Filterbank conv on zero-padded band: out[f,t] = sum_tau fb[f,tau]*band[t+tau-64]
// bandPad base pointer: element [t+kk] == band[t+kk-64] with zero halo.
__global__ void fbconv_wmma_kernel(const float* __restrict__ bandPad,
                                   const float* __restrict__ fb,
                                   float* __restrict__ out, int bs) {
    int lane = threadIdx.x & 31;
    int mr = lane & 15;
    int half = lane >> 4;
    int koff = half << 1;
    int t = blockIdx.x * 64 + mr;
    int f0 = blockIdx.y * 16;
    v8f acc[4] = {};
    for (int k0 = 0; k0 < 128; k0 += 4) {
        int kk = k0 + koff;
        v2f a;
        int o = f0 + mr;
        a[0] = fb[o * 128 + kk];
        a[1] = fb[o * 128 + kk + 1];
        const float* p = bandPad + t + kk;
#pragma unroll
        for (int nt = 0; nt < 4; ++nt) {
            v2f bb;
            bb[0] = p[nt * 16];
            bb[1] = p[nt * 16 + 1];
            acc[nt] = __builtin_amdgcn_wmma_f32_16x16x4_f32(false, a, false, bb, (short)0,
                                                            acc[nt], false, false);
        }
    }
#pragma unroll
    for (int nt = 0; nt < 4; ++nt)
        for (int r = 0; r < 8; ++r) {
            int fo = f0 + r + 8 * half;
            out[(size_t)fo * bs + t + nt * 16] = acc[nt][r];
        }
}

// 1x1 channel-mixing GEMM: out[o,t] = act( sum_i W[o,i]*X[i,t] + bias[o] (+res) )
// X/out/resid are pre-offset base pointers with per-channel strides.
__global__ void gemm_wmma_kernel(const float* __restrict__ X, const float* __restrict__ W,
                                 const float* __restrict__ bias,
                                 const float* __restrict__ resid,   // nullable
                                 float* __restrict__ out,
                                 int Kdim, int xStride, int oStride, int leaky) {
    int lane = threadIdx.x & 31;
    int mr = lane & 15;
    int half = lane >> 4;
    int koff = half << 1;
    int t = blockIdx.x * 64 + mr;
    int o0 = blockIdx.y * 16;
    v8f acc[4] = {};
    for (int k0 = 0; k0 < Kdim; k0 += 4) {
        int kk = k0 + koff;
        v2f a;
        int o = o0 + mr;
        a[0] = W[(size_t)o * Kdim + kk];
        a[1] = W[(size_t)o * Kdim + kk + 1];
        const float* x0 = X + (size_t)kk * xStride + t;
        const float* x1 = x0 + xStride;
        if (k0 + 4 < Kdim) {
            __builtin_prefetch(x0 + 4 * (size_t)xStride, 0, 1);
        }
#pragma unroll
        for (int nt = 0; nt < 4; ++nt) {
            v2f bb;
            bb[0] = x0[nt * 16];
            bb[1] = x1[nt * 16];
            acc[nt] = __builtin_amdgcn_wmma_f32_16x16x4_f32(false, a, false, bb, (short)0,
                                                            acc[nt], false, false);
        }
    }
#pragma unroll
    for (int nt = 0; nt < 4; ++nt)
        for (int r = 0; r < 8; ++r) {
            int o = o0 + r + 8 * half;
            float v = acc[nt][r] + bias[o];
            if (resid) v += resid[(size_t)o * oStride + t + nt * 16];
            if (leaky) v = (v > 0.f) ? v : 0.2f * v;
            out[(size_t)o * oStride + t + nt * 16] = v;
        }
}

// Dilated 3-tap conv as implicit GEMM with K=96 (kk = i*3 + d):
// out[o,t] = sum_i sum_d Wd[o,i,d] * X[i, t+(d-1)*dil] + bias[o]
// X/out are pre-offset padded base pointers (halo of 4 per channel).
__global__ void conv3_wmma_kernel(const float* __restrict__ X, const float* __restrict__ Wd,
                                  const float* __restrict__ bias, float* __restrict__ out,
                                  int stride, int dil) {
    int lane = threadIdx.x & 31;
    int mr = lane & 15;
    int half = lane >> 4;
    int koff = half << 1;
    int t = blockIdx.x * 64 + mr;
    int o0 = blockIdx.y * 16;
    v8f acc[4] = {};
    for (int k0 = 0; k0 < 96; k0 += 4) {
        int kk0 = k0 + koff;
        int kk1 = kk0 + 1;
        v2f a;
        int o = o0 + mr;
        a[0] = Wd[o * 96 + kk0];
        a[1] = Wd[o * 96 + kk1];
        int i0 = kk0 / 3, d0 = kk0 - 3 * i0;
        int i1 = kk1 / 3, d1 = kk1 - 3 * i1;
        const float* p0 = X + (size_t)i0 * stride + t + (d0 - 1) * dil;
        const float* p1 = X + (size_t)i1 * stride + t + (d1 - 1) * dil;
#pragma unroll
        for (int nt = 0; nt < 4; ++nt) {
            v2f bb;
            bb[0] = p0[nt * 16];
            bb[1] = p1[nt * 16];
            acc[nt] = __builtin_amdgcn_wmma_f32_16x16x4_f32(false, a, false, bb, (short)0,
                                                            acc[nt], false, false);
        }
    }
#pragma unroll
    for (int nt = 0; nt < 4; ++nt)
        for (int r = 0; r < 8; ++r) {
            int o = o0 + r + 8 * half;
            out[(size_t)o * stride + t + nt * 16] = acc[nt][r] + bias[o];
        }
}

// --------------------------- small helper kernels --------------------------

// Wcomb[a,i] = sum_m w_head[a,m]*w_out[m,i];  bcomb[a] = sum_m w_head[a,m]*b_out[m] + b_head[a]
__global__ void combine_weights_kernel(const float* __restrict__ w_head,
                                       const float* __restrict__ b_head,
                                       const float* __restrict__ w_out,
                                       const float* __restrict__ b_out,
                                       float* __restrict__ Wcomb, float* __restrict__ bcomb) {
    int a = blockIdx.x;     // 256
    int i = threadIdx.x;    // 32
    float s = 0.f;
    for (int m = 0; m < 256; ++m) s += w_head[a * 256 + m] * w_out[m * 32 + i];
    Wcomb[a * 32 + i] = s;
    if (i == 0) {
        float sb = 0.f;
        for (int m = 0; m < 256; ++m) sb += w_head[a * 256 + m] * b_out[m];
        bcomb[a] = sb + b_head[a];
    }
}

__global__ void atom_norm_kernel(const float* __restrict__ atoms, float* __restrict__ invnorm) {
    int a = blockIdx.x;
    float s = 0.f;
    for (int j = threadIdx.x; j < 512; j += blockDim.x) {
        float v = atoms[a * 512 + j];
        s += v * v;
    }
    __shared__ float sd[256];
    sd[threadIdx.x] = s; __syncthreads();
    for (int off = 128; off > 0; off >>= 1) {
        if (threadIdx.x < off) sd[threadIdx.x] += sd[threadIdx.x + off];
        __syncthreads();
    }
    if (threadIdx.x == 0) invnorm[a] = 1.f / (sqrtf(sd[0]) + 1e-12f);
}

// Per-slice argmax of c (single batch). Tie -> lowest flat index.
__global__ void slicemax_kernel(const float* __restrict__ c,
                                float* __restrict__ sMax, int* __restrict__ sArg,
                                int sliceLen) {
    int sl = blockIdx.x;
    int base = sl * sliceLen;
    float bv = -FLT_MAX; int bi = base;
    for (int j = threadIdx.x; j < sliceLen; j += blockDim.x) {
        int g = base + j;
        float v = c[g];
        if (v > bv) { bv = v; bi = g; }
    }
    __shared__ float sv[256]; __shared__ int si[256];
    sv[threadIdx.x] = bv; si[threadIdx.x] = bi;
    __syncthreads();
    for (int off = 128; off > 0; off >>= 1) {
        if (threadIdx.x < off) {
            float v2 = sv[threadIdx.x + off]; int i2 = si[threadIdx.x + off];
            if (v2 > sv[threadIdx.x] || (v2 == sv[threadIdx.x] && i2 < si[threadIdx.x])) {
                sv[threadIdx.x] = v2; si[threadIdx.x] = i2;
            }
        }
        __syncthreads();
    }
    if (threadIdx.x == 0) { sMax[sl] = sv[0]; sArg[sl] = si[0]; }
}

// Iterative top-k via slice tournament; then compute values at picked indices.
__global__ void topk_select_kernel(float* __restrict__ c,
                                   float* __restrict__ sMax, int* __restrict__ sArg,
                                   const float* __restrict__ x4, int xStride,
                                   const float* __restrict__ Wv, const float* __restrict__ bvp,
                                   int* __restrict__ aOut, int* __restrict__ sOut,
                                   float* __restrict__ vOut,
                                   int bs, int nSlices, int sliceLen, int kKeep) {
    __shared__ float sv[256]; __shared__ int ss[256];
    __shared__ int winSlice;
    for (int j = 0; j < kKeep; ++j) {
        // global best among slice maxima (tie -> smaller flat index = smaller slice)
        float bv = -FLT_MAX; int bsl = 0;
        for (int sl = threadIdx.x; sl < nSlices; sl += blockDim.x) {
            float v = sMax[sl];
            if (v > bv || (v == bv && sl < bsl)) { bv = v; bsl = sl; }
        }
        sv[threadIdx.x] = bv; ss[threadIdx.x] = bsl;
        __syncthreads();
        for (int off = 128; off > 0; off >>= 1) {
            if (threadIdx.x < off) {
                float v2 = sv[threadIdx.x + off]; int s2 = ss[threadIdx.x + off];
                if (v2 > sv[threadIdx.x] || (v2 == sv[threadIdx.x] && s2 < ss[threadIdx.x])) {
                    sv[threadIdx.x] = v2; ss[threadIdx.x] = s2;
                }
            }
            __syncthreads();
        }
        if (threadIdx.x == 0) {
            int sl = ss[0];
            int g = sArg[sl];
            winSlice = sl;
            aOut[j] = g / bs;          // bs is a power of two
            sOut[j] = g & (bs - 1);
            c[g] = -FLT_MAX;           // remove from further consideration
        }
        __syncthreads();
        // rescan winner slice
        int sl = winSlice;
        int base = sl * sliceLen;
        float lv = -FLT_MAX; int li = base;
        for (int q = threadIdx.x; q < sliceLen; q += blockDim.x) {
            int g = base + q;
            float v = c[g];
            if (v > lv) { lv = v; li = g; }
        }
        sv[threadIdx.x] = lv; ss[threadIdx.x] = li;
        __syncthreads();
        for (int off = 128; off > 0; off >>= 1) {
            if (threadIdx.x < off) {
                float v2 = sv[threadIdx.x + off]; int i2 = ss[threadIdx.x + off];
                if (v2 > sv[threadIdx.x] || (v2 == sv[threadIdx.x] && i2 < ss[threadIdx.x])) {
                    sv[threadIdx.x] = v2; ss[threadIdx.x] = i2;
                }
            }
            __syncthreads();
        }
        if (threadIdx.x == 0) { sMax[sl] = sv[0]; sArg[sl] = ss[0]; }
        __syncthreads();
    }
    // values at picked positions: v = Wv[a,:] . x4[:,s] + bvp[a]
    for (int j = threadIdx.x; j < kKeep; j += blockDim.x) {
        int a = aOut[j], sp = sOut[j];
        float s = bvp[a];
        for (int i = 0; i < 32; ++i) s += Wv[a * 32 + i] * x4[(size_t)i * xStride + sp];
        vOut[j] = s;
    }
}

// Deterministic transposed-conv reconstruction as gather:
// recon[t] = sum_j val_j * invnorm[a_j] * atoms[a_j, 256 + t - s_j], window |t-s|<=256
__global__ void recon_gather_kernel(const int* __restrict__ aArr, const int* __restrict__ sArr,
                                    const float* __restrict__ vArr,
                                    const float* __restrict__ atoms,
                                    const float* __restrict__ invnorm,
                                    float* __restrict__ recon_b, int bs, int kKeep) {
    int t = blockIdx.x * blockDim.x + threadIdx.x;
    if (t >= bs) return;
    float acc = 0.f;
    for (int j = 0; j < kKeep; ++j) {
        int d = t - sArr[j];
        if (d >= -256 && d < 256) {
            int a = aArr[j];
            acc += vArr[j] * invnorm[a] * atoms[a * 512 + 256 + d];
        }
    }
    recon_b[t] = acc;
}

// ------------------------------- host side ---------------------------------

struct BandP {
    const float *w_in, *b_in, *wd1, *bd1, *wc1, *bc1, *wd2, *bd2, *wc2, *bc2;
    const float *w_out, *b_out, *w_choice, *b_choice, *w_value, *b_value, *atoms;
};

static const int BAND_SIZES_H[6] = {32768, 16384, 8192, 4096, 2048, 1024};
static const int KEEP_H[6]       = {128, 128, 64, 64, 32, 32};

extern "C" void kernel_launch(void* const* d_in, const int* in_sizes, int n_in,
                              void* d_out, int out_size, void* d_ws, size_t ws_size,
                              hipStream_t stream) {
    // ---- map inputs (handle both insertion-order and sorted-key flattening) ----
    const float *x, *fb1, *fb2;
    BandP bp[6];
    bool insertion = (n_in >= 2 && in_sizes[1] == 16384);   // fb1 second => insertion order
    if (insertion) {
        x   = (const float*)d_in[0];
        fb1 = (const float*)d_in[1];
        fb2 = (const float*)d_in[2];
        for (int i = 0; i < 6; ++i) {
            int base = 3 + i * 17;
            bp[i].w_in     = (const float*)d_in[base + 0];
            bp[i].b_in     = (const float*)d_in[base + 1];
            bp[i].wd1      = (const float*)d_in[base + 2];
            bp[i].bd1      = (const float*)d_in[base + 3];
            bp[i].wc1      = (const float*)d_in[base + 4];
            bp[i].bc1      = (const float*)d_in[base + 5];
            bp[i].wd2      = (const float*)d_in[base + 6];
            bp[i].bd2      = (const float*)d_in[base + 7];
            bp[i].wc2      = (const float*)d_in[base + 8];
            bp[i].bc2      = (const float*)d_in[base + 9];
            bp[i].w_out    = (const float*)d_in[base + 10];
            bp[i].b_out    = (const float*)d_in[base + 11];
            bp[i].w_choice = (const float*)d_in[base + 12];
            bp[i].b_choice = (const float*)d_in[base + 13];
            bp[i].w_value  = (const float*)d_in[base + 14];
            bp[i].b_value  = (const float*)d_in[base + 15];
            bp[i].atoms    = (const float*)d_in[base + 16];
        }
    } else {
        // sorted keys: bands '1024','16384','2048','32768','4096','8192', leaves sorted:
        // atoms,b_choice,b_in,b_out,b_value,bc1,bc2,bd1,bd2,w_choice,w_in,w_out,w_value,wc1,wc2,wd1,wd2
        static const int mapping[6] = {3, 1, 5, 4, 2, 0};   // BAND_SIZES order -> sorted block
        for (int i = 0; i < 6; ++i) {
            int base = mapping[i] * 17;
            bp[i].atoms    = (const float*)d_in[base + 0];
            bp[i].b_choice = (const float*)d_in[base + 1];
            bp[i].b_in     = (const float*)d_in[base + 2];
            bp[i].b_out    = (const float*)d_in[base + 3];
            bp[i].b_value  = (const float*)d_in[base + 4];
            bp[i].bc1      = (const float*)d_in[base + 5];
            bp[i].bc2      = (const float*)d_in[base + 6];
            bp[i].bd1      = (const float*)d_in[base + 7];
            bp[i].bd2      = (const float*)d_in[base + 8];
            bp[i].w_choice = (const float*)d_in[base + 9];
            bp[i].w_in     = (const float*)d_in[base + 10];
            bp[i].w_out    = (const float*)d_in[base + 11];
            bp[i].w_value  = (const float*)d_in[base + 12];
            bp[i].wc1      = (const float*)d_in[base + 13];
            bp[i].wc2      = (const float*)d_in[base + 14];
            bp[i].wd1      = (const float*)d_in[base + 15];
            bp[i].wd2      = (const float*)d_in[base + 16];
        }
        fb1 = (const float*)d_in[102];
        fb2 = (const float*)d_in[103];
        x   = (const float*)d_in[104];
    }

    // ---- workspace layout (~65 MB of floats) ----
    float* w = (float*)d_ws;
    size_t off = 0;
    auto alloc = [&](size_t n) { float* p = w + off; off += n; return p; };
    float* Cre     = alloc((size_t)BATCH * NFREQ);
    float* Cim     = alloc((size_t)BATCH * NFREQ);
    float* Tre     = alloc((size_t)BATCH * NFREQ);
    float* Tim     = alloc((size_t)BATCH * NFREQ);
    float* bandPad = alloc((size_t)BATCH * (NSAMP + 128));  // 64-elem zero halo
    float* recon   = alloc((size_t)BATCH * NSAMP);
    float* buf128  = alloc((size_t)128 * NSAMP);            // per-batch, stride bs
    float* buf32a  = alloc((size_t)32 * (NSAMP + 8));       // per-batch, stride bs+8, halo 4
    float* buf32b  = alloc((size_t)32 * (NSAMP + 8));
    float* buf32c  = alloc((size_t)32 * (NSAMP + 8));
    float* bufC    = alloc((size_t)256 * NSAMP);            // combined choice logits
    float* Wc      = alloc(256 * 32);
    float* bcp     = alloc(256);
    float* Wv      = alloc(256 * 32);
    float* bvp     = alloc(256);
    float* invnorm = alloc(256);
    float* sMax    = alloc(2048);
    int*   sArg    = (int*)alloc(2048);
    int*   aArr    = (int*)alloc(128);
    int*   sArr    = (int*)alloc(128);
    float* vArr    = alloc(128);
    (void)ws_size; (void)out_size;

    // ---- forward spectrum of x ----
    dft_forward_kernel<<<dim3(NFREQ, BATCH), 256, 0, stream>>>(x, Cre, Cim);

    // ---- per band ----
    for (int ib = 0; ib < 6; ++ib) {
        const int bs = BAND_SIZES_H[ib];
        const int bsp = bs + 8;
        const int kKeep = KEEP_H[ib];
        const int kstart = (bs == 1024) ? 0 : bs / 4;
        const int kstop = bs / 2;
        const float* fb = (ib == 0) ? fb1 : fb2;
        const BandP& P = bp[ib];

        zero_band_pad_kernel<<<BATCH, 128, 0, stream>>>(bandPad, bs);
        zero_ch_pad_kernel<<<1, 256, 0, stream>>>(buf32a, bs);
        zero_ch_pad_kernel<<<1, 256, 0, stream>>>(buf32b, bs);
        zero_ch_pad_kernel<<<1, 256, 0, stream>>>(buf32c, bs);

        band_synth_kernel<<<dim3(bs, BATCH), 256, 0, stream>>>(Cre, Cim, bandPad, bs, kstart, kstop);

        // fold w_out into the choice/value heads (256x32 combined weights)
        combine_weights_kernel<<<256, 32, 0, stream>>>(P.w_choice, P.b_choice, P.w_out, P.b_out, Wc, bcp);
        combine_weights_kernel<<<256, 32, 0, stream>>>(P.w_value,  P.b_value,  P.w_out, P.b_out, Wv, bvp);
        atom_norm_kernel<<<256, 256, 0, stream>>>(P.atoms, invnorm);

        const int L = 256 * bs;
        const int sliceLen = 4096;
        const int nSlices = L / sliceLen;

        for (int b = 0; b < BATCH; ++b) {
            const float* bandb = bandPad + (size_t)b * (bs + 128);

            fbconv_wmma_kernel<<<dim3(bs / 64, 8), 32, 0, stream>>>(bandb, fb, buf128, bs);
            gemm_wmma_kernel<<<dim3(bs / 64, 2), 32, 0, stream>>>(buf128, P.w_in, P.b_in,
                                                                  nullptr, buf32a + 4,
                                                                  128, bs, bsp, 0);
            conv3_wmma_kernel<<<dim3(bs / 64, 2), 32, 0, stream>>>(buf32a + 4, P.wd1, P.bd1,
                                                                   buf32b + 4, bsp, 1);
            gemm_wmma_kernel<<<dim3(bs / 64, 2), 32, 0, stream>>>(buf32b + 4, P.wc1, P.bc1,
                                                                  buf32a + 4, buf32c + 4,
                                                                  32, bsp, bsp, 1);
            conv3_wmma_kernel<<<dim3(bs / 64, 2), 32, 0, stream>>>(buf32c + 4, P.wd2, P.bd2,
                                                                   buf32b + 4, bsp, 3);
            gemm_wmma_kernel<<<dim3(bs / 64, 2), 32, 0, stream>>>(buf32b + 4, P.wc2, P.bc2,
                                                                  buf32c + 4, buf32a + 4,
                                                                  32, bsp, bsp, 1);
            // combined choice logits (no softmax needed: argsort-invariant)
            gemm_wmma_kernel<<<dim3(bs / 64, 16), 32, 0, stream>>>(buf32a + 4, Wc, bcp,
                                                                   nullptr, bufC,
                                                                   32, bsp, bs, 0);
            slicemax_kernel<<<nSlices, 256, 0, stream>>>(bufC, sMax, sArg, sliceLen);
            topk_select_kernel<<<1, 256, 0, stream>>>(bufC, sMax, sArg, buf32a + 4, bsp,
                                                      Wv, bvp, aArr, sArr, vArr,
                                                      bs, nSlices, sliceLen, kKeep);
            recon_gather_kernel<<<(bs + 255) / 256, 256, 0, stream>>>(
                aArr, sArr, vArr, P.atoms, invnorm, recon + (size_t)b * bs, bs, kKeep);
        }

        band_dft_kernel<<<dim3(kstop - kstart, BATCH), 256, 0, stream>>>(recon, Tre, Tim, bs, kstart);
    }

    // ---- recombine to output ----
    final_idft_kernel<<<dim3(NSAMP, BATCH), 256, 0, stream>>>(Tre, Tim, (float*)d_out);
}